// GNNEncoder_12515534701155
// MI455X (gfx1250) — compile-verified
//
#include <hip/hip_runtime.h>

#define N_NODES   100000
#define N_EDGES   1200000
#define N_GRAPHS  500
#define IN_DIM    128
#define HID       64
#define OUT_DIM   16

typedef __attribute__((ext_vector_type(2))) float v2f;
typedef __attribute__((ext_vector_type(8))) float v8f;

static inline int cdiv(int a, int b) { return (a + b - 1) / b; }

// ---------------- degree / normalization ----------------
__global__ void k_set_one(float* __restrict__ p, int n) {
    int i = blockIdx.x * blockDim.x + threadIdx.x;
    if (i < n) p[i] = 1.0f;
}

__global__ void k_deg_count(const int* __restrict__ dst, float* __restrict__ deg, int n_edges) {
    int e = blockIdx.x * blockDim.x + threadIdx.x;
    if (e < n_edges) atomicAdd(&deg[dst[e]], 1.0f);
}

__global__ void k_deg_to_dis(float* __restrict__ deg, int n) {
    int i = blockIdx.x * blockDim.x + threadIdx.x;
    if (i < n) deg[i] = rsqrtf(deg[i]);   // deg >= 1, always finite
}

// ---------------- GEMM via V_WMMA_F32_16X16X4_F32 ----------------
// One wave computes a full 16 x FOUT strip of H = X @ W:  A fragments are
// loaded once per k-step and reused across FOUT/16 column tiles.
// Compile-time K/FOUT -> all loads are base + immediate offset, full unroll.
// A layout (16x4 f32): lanes 0-15 hold K={k,k+1}, lanes 16-31 hold K={k+2,k+3}.
// B layout (4x16 f32): VGPR v holds row K=k+v (lanes 0-15) / K=k+v+2 (lanes 16-31).
// C/D layout: VGPR r -> row r (lanes 0-15) / row r+8 (lanes 16-31), col = lane%16.
template <int K, int FOUT, bool RELU>
__global__ void k_gemm_wmma(const float* __restrict__ X, const float* __restrict__ W,
                            float* __restrict__ H) {
    constexpr int NT = FOUT / 16;     // column tiles per wave
    const int lane = threadIdx.x;     // 0..31, one full wave; EXEC all ones
    const int l16  = lane & 15;
    const int kh   = lane >> 4;       // K half-select (0 or 1)
    const int row  = blockIdx.x * 16 + l16;

    const float* __restrict__ ap = X + (size_t)row * K + kh * 2;        // A: +k imm offsets
    const float* __restrict__ bp = W + (size_t)(kh * 2) * FOUT + l16;   // B: +k*FOUT imm offsets

    v8f acc[NT];
#pragma unroll
    for (int t = 0; t < NT; ++t) acc[t] = (v8f){};

#pragma unroll
    for (int k = 0; k < K; k += 4) {
        float a0 = ap[k];           // contiguous pair -> global_load_b64 offset:k*4
        float a1 = ap[k + 1];
        if (RELU) { a0 = fmaxf(a0, 0.0f); a1 = fmaxf(a1, 0.0f); }
        v2f a;
        a[0] = a0;
        a[1] = a1;
#pragma unroll
        for (int t = 0; t < NT; ++t) {
            v2f b;
            b[0] = bp[(size_t)k * FOUT + t * 16];
            b[1] = bp[(size_t)(k + 1) * FOUT + t * 16];
            acc[t] = __builtin_amdgcn_wmma_f32_16x16x4_f32(
                /*neg_a=*/false, a, /*neg_b=*/false, b,
                /*c_mod=*/(short)0, acc[t], /*reuse_a=*/false, /*reuse_b=*/false);
        }
    }

    const int orow = blockIdx.x * 16 + kh * 8;
#pragma unroll
    for (int t = 0; t < NT; ++t) {
        float* __restrict__ hp = H + (size_t)orow * FOUT + t * 16 + l16;
#pragma unroll
        for (int r = 0; r < 8; ++r) hp[(size_t)r * FOUT] = acc[t][r];
    }
}

// ---------------- aggregation ----------------
// agg = h * dis^2 + bias   (self-loop term; also serves as the accumulator init)
__global__ void k_init_agg(const float* __restrict__ H, const float* __restrict__ dis,
                           const float* __restrict__ bias, float* __restrict__ agg,
                           int n_nodes, int F) {
    int t = blockIdx.x * blockDim.x + threadIdx.x;
    int total = n_nodes * F;
    if (t < total) {
        int i = t / F;
        int f = t - i * F;
        float d = dis[i];
        agg[t] = H[t] * d * d + bias[f];
    }
}

// atomicAdd(agg[dst], h[src] * dis[src]*dis[dst]); one thread per (edge, 4-float chunk)
template <int F>
__global__ void k_edge_scatter(const float* __restrict__ H, const int* __restrict__ src,
                               const int* __restrict__ dst, const float* __restrict__ dis,
                               float* __restrict__ agg, int n_edges) {
    constexpr int CHUNKS = F / 4;     // power of two -> shift/mask indexing
    int t = blockIdx.x * blockDim.x + threadIdx.x;
    int total = n_edges * CHUNKS;
    if (t >= total) return;
    int e = t / CHUNKS;
    int c = t & (CHUNKS - 1);
    int s = src[e];
    int d = dst[e];
    float norm = dis[s] * dis[d];
    float4 v = ((const float4*)(H + (size_t)s * F))[c];   // global_load_b128
    float* ap = agg + (size_t)d * F + c * 4;
    atomicAdd(ap + 0, v.x * norm);
    atomicAdd(ap + 1, v.y * norm);
    atomicAdd(ap + 2, v.z * norm);
    atomicAdd(ap + 3, v.w * norm);
}

// ---------------- pooling ----------------
__global__ void k_zero(float* __restrict__ p, int n) {
    int i = blockIdx.x * blockDim.x + threadIdx.x;
    if (i < n) p[i] = 0.0f;
}

__global__ void k_pool_scatter(const float* __restrict__ h3, const int* __restrict__ batch,
                               float* __restrict__ sums, float* __restrict__ counts,
                               int n_nodes) {
    int t = blockIdx.x * blockDim.x + threadIdx.x;
    int total = n_nodes * OUT_DIM;
    if (t < total) {
        int i = t / OUT_DIM;
        int f = t & (OUT_DIM - 1);
        int g = batch[i];
        atomicAdd(&sums[g * OUT_DIM + f], h3[t]);
        if (f == 0) atomicAdd(&counts[g], 1.0f);
    }
}

__global__ void k_pool_final(const float* __restrict__ sums, const float* __restrict__ counts,
                             float* __restrict__ out) {
    int t = blockIdx.x * blockDim.x + threadIdx.x;
    if (t < N_GRAPHS * OUT_DIM) {
        int g = t / OUT_DIM;
        out[t] = sums[t] / fmaxf(counts[g], 1.0f);
    }
}

// ---------------- driver ----------------
extern "C" void kernel_launch(void* const* d_in, const int* in_sizes, int n_in,
                              void* d_out, int out_size, void* d_ws, size_t ws_size,
                              hipStream_t stream) {
    const float* x     = (const float*)d_in[0];
    const int*   eidx  = (const int*)d_in[1];     // shape (2, E): row0=src, row1=dst
    const int*   batch = (const int*)d_in[2];
    const float* W1 = (const float*)d_in[3];
    const float* b1 = (const float*)d_in[4];
    const float* W2 = (const float*)d_in[5];
    const float* b2 = (const float*)d_in[6];
    const float* W3 = (const float*)d_in[7];
    const float* b3 = (const float*)d_in[8];
    float* out = (float*)d_out;

    const int* src = eidx;
    const int* dst = eidx + N_EDGES;

    // workspace layout (floats), all offsets 64-float (256 B) aligned
    float* ws   = (float*)d_ws;
    float* dis  = ws;                                  // N_NODES (deg, then rsqrt in place)
    float* h    = ws + 100032;                         // N_NODES * HID
    float* agg  = h + (size_t)N_NODES * HID;           // N_NODES * HID
    float* sums = agg + (size_t)N_NODES * HID;         // N_GRAPHS * OUT_DIM
    float* cnts = sums + N_GRAPHS * OUT_DIM;           // N_GRAPHS

    const int B = 256;
    const dim3 blk(B);

    // degree + dis = (1 + indeg)^-1/2
    k_set_one<<<cdiv(N_NODES, B), blk, 0, stream>>>(dis, N_NODES);
    k_deg_count<<<cdiv(N_EDGES, B), blk, 0, stream>>>(dst, dis, N_EDGES);
    k_deg_to_dis<<<cdiv(N_NODES, B), blk, 0, stream>>>(dis, N_NODES);

    const int MT = N_NODES / 16;   // 6250, exact (100000 = 16 * 6250)

    // ---- layer 1: 128 -> 64 ----
    k_gemm_wmma<IN_DIM, HID, false><<<dim3(MT), dim3(32), 0, stream>>>(x, W1, h);
    k_init_agg<<<cdiv(N_NODES * HID, B), blk, 0, stream>>>(h, dis, b1, agg, N_NODES, HID);
    k_edge_scatter<HID><<<cdiv(N_EDGES * (HID / 4), B), blk, 0, stream>>>(h, src, dst, dis, agg, N_EDGES);

    // ---- layer 2: 64 -> 64 (relu folded into A-load) ----
    k_gemm_wmma<HID, HID, true><<<dim3(MT), dim3(32), 0, stream>>>(agg, W2, h);
    k_init_agg<<<cdiv(N_NODES * HID, B), blk, 0, stream>>>(h, dis, b2, agg, N_NODES, HID);
    k_edge_scatter<HID><<<cdiv(N_EDGES * (HID / 4), B), blk, 0, stream>>>(h, src, dst, dis, agg, N_EDGES);

    // ---- layer 3: 64 -> 16 (reuse h as h3, agg as agg3; stride OUT_DIM) ----
    k_gemm_wmma<HID, OUT_DIM, true><<<dim3(MT), dim3(32), 0, stream>>>(agg, W3, h);
    k_init_agg<<<cdiv(N_NODES * OUT_DIM, B), blk, 0, stream>>>(h, dis, b3, agg, N_NODES, OUT_DIM);
    k_edge_scatter<OUT_DIM><<<cdiv(N_EDGES * (OUT_DIM / 4), B), blk, 0, stream>>>(h, src, dst, dis, agg, N_EDGES);

    // ---- global mean pool ----
    k_zero<<<cdiv(N_GRAPHS * OUT_DIM + N_GRAPHS, B), blk, 0, stream>>>(sums, N_GRAPHS * OUT_DIM + N_GRAPHS);
    k_pool_scatter<<<cdiv(N_NODES * OUT_DIM, B), blk, 0, stream>>>(agg, batch, sums, cnts, N_NODES);
    k_pool_final<<<cdiv(N_GRAPHS * OUT_DIM, B), blk, 0, stream>>>(sums, cnts, out);
}